// SparseGraphAttentionLayer_36816459661413
// MI455X (gfx1250) — compile-verified
//
#include <hip/hip_runtime.h>

#define N_NODES 50000
#define E_EDGES 1600000
#define ET_EDGES (E_EDGES + N_NODES)   // edges + self loops = 1650000
#define IN_FEAT 128
#define OUT_FEAT 64
#define NEG_SLOPE 0.2f

typedef __attribute__((ext_vector_type(2))) float v2f;
typedef __attribute__((ext_vector_type(8))) float v8f;

// ---------------------------------------------------------------------------
// Kernel 1: Wh = x @ W via V_WMMA_F32_16X16X4_F32 (fp32, exact).
// One wave computes one 16x16 output tile. Block = 128 threads = 4 waves,
// covering the full OUT_FEAT=64 width. grid.x = 50000/16 = 3125 M-tiles.
// ---------------------------------------------------------------------------
__global__ __launch_bounds__(128)
void gat_gemm_wmma(const float* __restrict__ x, const float* __restrict__ W,
                   float* __restrict__ Wh) {
    const int wave = threadIdx.x >> 5;        // 0..3 -> N-tile
    const int lane = threadIdx.x & 31;
    const int m0   = blockIdx.x * 16;
    const int n0   = wave * 16;
    const int mrow = m0 + (lane & 15);        // A row for this lane
    const int kl   = (lane < 16) ? 0 : 2;     // K sub-offset per half-wave
    const int ncol = n0 + (lane & 15);        // B/C column for this lane

    const float* __restrict__ arow = x + (size_t)mrow * IN_FEAT;

    v8f c = {};
    #pragma unroll 4
    for (int k0 = 0; k0 < IN_FEAT; k0 += 4) {
        // A fragment: lane holds A[mrow][k0+kl .. k0+kl+1] (contiguous pair)
        v2f a;
        a.x = arow[k0 + kl + 0];
        a.y = arow[k0 + kl + 1];
        // B fragment: VGPR v holds B[k0+kl+v][ncol]
        v2f b;
        b.x = W[(size_t)(k0 + kl + 0) * OUT_FEAT + ncol];
        b.y = W[(size_t)(k0 + kl + 1) * OUT_FEAT + ncol];
        c = __builtin_amdgcn_wmma_f32_16x16x4_f32(
                /*neg_a=*/false, a, /*neg_b=*/false, b,
                /*c_mod=*/(short)0, c, /*reuse_a=*/false, /*reuse_b=*/false);
    }

    // C layout: VGPR v -> row m0+v (lanes 0-15) / m0+v+8 (lanes 16-31)
    const int rbase = m0 + ((lane < 16) ? 0 : 8);
    #pragma unroll
    for (int v = 0; v < 8; ++v) {
        Wh[(size_t)(rbase + v) * OUT_FEAT + ncol] = c[v];
    }
}

// ---------------------------------------------------------------------------
// Kernel 2: per-node attention scores  s_src[i] = Wh[i]·a[:64],
//                                      s_dst[i] = Wh[i]·a[64:]
// One wave per node, lane covers features {lane, lane+32}, shuffle-reduce.
// ---------------------------------------------------------------------------
__global__ __launch_bounds__(256)
void gat_scores(const float* __restrict__ Wh, const float* __restrict__ a,
                float* __restrict__ s_src, float* __restrict__ s_dst) {
    const int node = (int)((blockIdx.x * (unsigned)blockDim.x + threadIdx.x) >> 5);
    const int lane = threadIdx.x & 31;
    if (node >= N_NODES) return;

    const float w0 = Wh[(size_t)node * OUT_FEAT + lane];
    const float w1 = Wh[(size_t)node * OUT_FEAT + lane + 32];
    float ss = w0 * a[lane]      + w1 * a[lane + 32];
    float sd = w0 * a[64 + lane] + w1 * a[96 + lane];
    #pragma unroll
    for (int off = 16; off > 0; off >>= 1) {
        ss += __shfl_down(ss, off, 32);
        sd += __shfl_down(sd, off, 32);
    }
    if (lane == 0) {
        s_src[node] = ss;
        s_dst[node] = sd;
    }
}

// Order-preserving float <-> uint mapping for atomic max on signed floats.
__device__ __forceinline__ unsigned fkey(float f) {
    unsigned u = __float_as_uint(f);
    return (u & 0x80000000u) ? ~u : (u | 0x80000000u);
}
__device__ __forceinline__ float funkey(unsigned k) {
    unsigned u = (k & 0x80000000u) ? (k ^ 0x80000000u) : ~k;
    return __uint_as_float(u);
}

__device__ __forceinline__ void edge_endpoints(const int* __restrict__ ei,
                                               int e, int& src, int& dst) {
    if (e < E_EDGES) {
        src = ei[e];
        dst = ei[E_EDGES + e];
    } else {
        src = dst = e - E_EDGES;      // self loop
    }
}

__device__ __forceinline__ float edge_score(const float* __restrict__ s_src,
                                            const float* __restrict__ s_dst,
                                            int src, int dst) {
    float ev = s_src[src] + s_dst[dst];
    return (ev > 0.0f) ? ev : NEG_SLOPE * ev;
}

// ---------------------------------------------------------------------------
// Kernel 3: segment max of leaky-relu scores via u32 atomic max on keys.
// e_max_key must be zero-initialized (key 0 < key(-inf)).
// ---------------------------------------------------------------------------
__global__ __launch_bounds__(256)
void gat_edge_max(const int* __restrict__ ei,
                  const float* __restrict__ s_src, const float* __restrict__ s_dst,
                  unsigned* __restrict__ e_max_key) {
    const int e = (int)(blockIdx.x * (unsigned)blockDim.x + threadIdx.x);
    if (e >= ET_EDGES) return;
    int src, dst;
    edge_endpoints(ei, e, src, dst);
    const float ev = edge_score(s_src, s_dst, src, dst);
    atomicMax(&e_max_key[dst], fkey(ev));
}

// ---------------------------------------------------------------------------
// Kernel 4: denom[dst] += exp(e - e_max[dst])   (denom zero-initialized)
// ---------------------------------------------------------------------------
__global__ __launch_bounds__(256)
void gat_edge_denom(const int* __restrict__ ei,
                    const float* __restrict__ s_src, const float* __restrict__ s_dst,
                    const unsigned* __restrict__ e_max_key,
                    float* __restrict__ denom) {
    const int e = (int)(blockIdx.x * (unsigned)blockDim.x + threadIdx.x);
    if (e >= ET_EDGES) return;
    int src, dst;
    edge_endpoints(ei, e, src, dst);
    const float ev = edge_score(s_src, s_dst, src, dst);
    const float ex = __expf(ev - funkey(e_max_key[dst]));
    atomicAdd(&denom[dst], ex);
}

// ---------------------------------------------------------------------------
// Kernel 5: out[dst] += alpha * Wh[src]. One wave per edge, lane handles
// features {lane, lane+32}. out must be zero-initialized.
// ---------------------------------------------------------------------------
__global__ __launch_bounds__(256)
void gat_edge_scatter(const int* __restrict__ ei,
                      const float* __restrict__ s_src, const float* __restrict__ s_dst,
                      const unsigned* __restrict__ e_max_key,
                      const float* __restrict__ denom,
                      const float* __restrict__ Wh,
                      float* __restrict__ out) {
    const long long t = (long long)blockIdx.x * blockDim.x + threadIdx.x;
    const int e = (int)(t >> 5);
    const int lane = threadIdx.x & 31;
    if (e >= ET_EDGES) return;
    int src, dst;
    edge_endpoints(ei, e, src, dst);
    const float ev = edge_score(s_src, s_dst, src, dst);   // uniform across wave
    const float alpha = __expf(ev - funkey(e_max_key[dst])) / denom[dst];

    const float w0 = Wh[(size_t)src * OUT_FEAT + lane];
    const float w1 = Wh[(size_t)src * OUT_FEAT + lane + 32];
    atomicAdd(&out[(size_t)dst * OUT_FEAT + lane],      w0 * alpha);
    atomicAdd(&out[(size_t)dst * OUT_FEAT + lane + 32], w1 * alpha);
}

// ---------------------------------------------------------------------------
// Kernel 6: ELU in place.
// ---------------------------------------------------------------------------
__global__ __launch_bounds__(256)
void gat_elu(float* __restrict__ out, int n) {
    const int i = (int)(blockIdx.x * (unsigned)blockDim.x + threadIdx.x);
    if (i >= n) return;
    const float h = out[i];
    out[i] = (h > 0.0f) ? h : (__expf(h) - 1.0f);
}

// ---------------------------------------------------------------------------
extern "C" void kernel_launch(void* const* d_in, const int* in_sizes, int n_in,
                              void* d_out, int out_size, void* d_ws, size_t ws_size,
                              hipStream_t stream) {
    const float* x  = (const float*)d_in[0];       // N x 128
    const int*   ei = (const int*)d_in[1];         // 2 x E
    const float* W  = (const float*)d_in[2];       // 128 x 64
    const float* a  = (const float*)d_in[3];       // 128 x 1

    // Workspace layout
    char* ws = (char*)d_ws;
    float*    Wh        = (float*)ws;                                   // N*64
    float*    s_src     = (float*)(ws + (size_t)N_NODES * OUT_FEAT * 4);
    float*    s_dst     = s_src + N_NODES;
    unsigned* e_max_key = (unsigned*)(s_dst + N_NODES);
    float*    denom     = (float*)(e_max_key + N_NODES);
    float*    out       = (float*)d_out;                                // N*64

    // Zero-init accumulators (capturable async memsets)
    hipMemsetAsync(e_max_key, 0, (size_t)N_NODES * sizeof(unsigned), stream);
    hipMemsetAsync(denom,     0, (size_t)N_NODES * sizeof(float),    stream);
    hipMemsetAsync(out,       0, (size_t)out_size * sizeof(float),   stream);

    // 1. Wh = x @ W  (WMMA fp32)
    gat_gemm_wmma<<<N_NODES / 16, 128, 0, stream>>>(x, W, Wh);

    // 2. attention scores
    gat_scores<<<(N_NODES * 32 + 255) / 256, 256, 0, stream>>>(Wh, a, s_src, s_dst);

    // 3. segment max
    gat_edge_max<<<(ET_EDGES + 255) / 256, 256, 0, stream>>>(ei, s_src, s_dst, e_max_key);

    // 4. softmax denominator
    gat_edge_denom<<<(ET_EDGES + 255) / 256, 256, 0, stream>>>(ei, s_src, s_dst, e_max_key, denom);

    // 5. weighted scatter-add (wave per edge)
    gat_edge_scatter<<<((long long)ET_EDGES * 32 + 255) / 256, 256, 0, stream>>>(
        ei, s_src, s_dst, e_max_key, denom, Wh, out);

    // 6. ELU
    gat_elu<<<(N_NODES * OUT_FEAT + 255) / 256, 256, 0, stream>>>(out, N_NODES * OUT_FEAT);
}